// ASRmodel_bg_56977036148942
// MI455X (gfx1250) — compile-verified
//
#include <hip/hip_runtime.h>
#include <math.h>

// ---------------------------------------------------------------------------
// ASR angular-spectrum model for MI455X (gfx1250, wave32).
//
//   out = ifft2( (fft2(H*W) - B) * conj(T) ),  T = exp(i*(z/C_LAMBDA)*Q)
//
// 1024-point complex FFTs are done per-wave with the four-step algorithm:
// radix-32 DFT stages as 32x32x32 fp32 matmuls on V_WMMA_F32_16X16X4_F32.
// ---------------------------------------------------------------------------

typedef __attribute__((ext_vector_type(2))) float v2f;
typedef __attribute__((ext_vector_type(8))) float v8f;

#define PI_F 3.14159265358979323846f
#define C_LAMBDA_F 6.37e-07f

static __device__ __forceinline__ v8f wmma_f32(v2f a, v2f b, v8f c) {
  // D(16x16,f32) = A(16x4,f32) * B(4x16,f32) + C
  // args: (neg_a, A, neg_b, B, c_mod, C, reuse_a, reuse_b)
  return __builtin_amdgcn_wmma_f32_16x16x4_f32(false, a, false, b, (short)0, c,
                                               false, false);
}

// ---------------------------------------------------------------------------
// In-wave 1024-point complex FFT (decimation N = 32*32).
//   re/im : this wave's LDS planes, x[t] at re[t]/im[t], t = n1 + 32*n2
//   Cm/Sm : LDS 32x32 tables, Cm[k*32+n] = cos(2pi*k*n/32), Sm = sin(...)
//   dir   : +1 forward (e^{-i}), -1 inverse (e^{+i})
//   scale : applied to final values
// Result left in re/im in natural order.
// ---------------------------------------------------------------------------
static __device__ __forceinline__ void wave_fft1024(
    float* re, float* im, const float* Cm, const float* Sm,
    float dir, float scale, int lane)
{
  const int h   = lane >> 4;   // lane half (k/row split per WMMA layout)
  const int l16 = lane & 15;

  v8f cre[2][2], cim[2][2];

  // ---------------- Stage 1: Y = D * F, D[m][k] = x[m + 32k] --------------
#pragma unroll
  for (int i = 0; i < 2; ++i)
#pragma unroll
    for (int j = 0; j < 2; ++j) {
      cre[i][j] = (v8f){0.f,0.f,0.f,0.f,0.f,0.f,0.f,0.f};
      cim[i][j] = (v8f){0.f,0.f,0.f,0.f,0.f,0.f,0.f,0.f};
    }

#pragma unroll
  for (int kb = 0; kb < 8; ++kb) {
    const int k0 = 4 * kb + 2 * h;          // A/B layout: reg r holds K = k0+r
    v2f aRe[2], aIm[2], bC[2], bSp[2], bSm[2];
#pragma unroll
    for (int i = 0; i < 2; ++i) {
      const int m = l16 + 16 * i;
      aRe[i].x = re[m + 32 * k0];      aRe[i].y = re[m + 32 * (k0 + 1)];
      aIm[i].x = im[m + 32 * k0];      aIm[i].y = im[m + 32 * (k0 + 1)];
    }
#pragma unroll
    for (int j = 0; j < 2; ++j) {
      const int nn = l16 + 16 * j;
      v2f c, s;
      c.x = Cm[k0 * 32 + nn];          c.y = Cm[(k0 + 1) * 32 + nn];
      s.x = Sm[k0 * 32 + nn];          s.y = Sm[(k0 + 1) * 32 + nn];
      bC[j] = c;
      bSp[j].x = s.x * dir;            bSp[j].y = s.y * dir;   //  dir*S
      bSm[j].x = -bSp[j].x;            bSm[j].y = -bSp[j].y;   // -dir*S
    }
    // F = C - i*dir*S :  Yre = Dre*C + dir*Dim*S ; Yim = Dim*C - dir*Dre*S
#pragma unroll
    for (int i = 0; i < 2; ++i)
#pragma unroll
      for (int j = 0; j < 2; ++j) {
        cre[i][j] = wmma_f32(aRe[i], bC[j],  cre[i][j]);
        cre[i][j] = wmma_f32(aIm[i], bSp[j], cre[i][j]);
        cim[i][j] = wmma_f32(aIm[i], bC[j],  cim[i][j]);
        cim[i][j] = wmma_f32(aRe[i], bSm[j], cim[i][j]);
      }
  }

  // ------------- Twiddle w = e^{-i*dir*2pi*m*n/1024}; Y' -> LDS -----------
#pragma unroll
  for (int i = 0; i < 2; ++i)
#pragma unroll
    for (int j = 0; j < 2; ++j)
#pragma unroll
      for (int r = 0; r < 8; ++r) {
        const int m = 16 * i + r + 8 * h;   // C/D layout: reg r, lane half h
        const int n = 16 * j + l16;
        const int t = (m * n) & 1023;
        const float ang = -dir * (2.0f * PI_F / 1024.0f) * (float)t;
        float sv, cv;
        __sincosf(ang, &sv, &cv);
        const float yr = cre[i][j][r], yi = cim[i][j][r];
        re[m * 32 + n] = yr * cv - yi * sv;  // Y'[m][n] at [m*32+n]
        im[m * 32 + n] = yr * sv + yi * cv;
      }

  // ---------------- Stage 2: X = F * Y' -----------------------------------
#pragma unroll
  for (int i = 0; i < 2; ++i)
#pragma unroll
    for (int j = 0; j < 2; ++j) {
      cre[i][j] = (v8f){0.f,0.f,0.f,0.f,0.f,0.f,0.f,0.f};
      cim[i][j] = (v8f){0.f,0.f,0.f,0.f,0.f,0.f,0.f,0.f};
    }

#pragma unroll
  for (int kb = 0; kb < 8; ++kb) {
    const int k0 = 4 * kb + 2 * h;
    v2f aC[2], aSp[2], aSm[2], bRe[2], bIm[2];
#pragma unroll
    for (int i = 0; i < 2; ++i) {
      const int m = l16 + 16 * i;          // row of F
      v2f c, s;
      c.x = Cm[m * 32 + k0];               c.y = Cm[m * 32 + k0 + 1];
      s.x = Sm[m * 32 + k0];               s.y = Sm[m * 32 + k0 + 1];
      aC[i] = c;
      aSp[i].x = s.x * dir;                aSp[i].y = s.y * dir;
      aSm[i].x = -aSp[i].x;                aSm[i].y = -aSp[i].y;
    }
#pragma unroll
    for (int j = 0; j < 2; ++j) {
      const int nn = l16 + 16 * j;
      bRe[j].x = re[k0 * 32 + nn];         bRe[j].y = re[(k0 + 1) * 32 + nn];
      bIm[j].x = im[k0 * 32 + nn];         bIm[j].y = im[(k0 + 1) * 32 + nn];
    }
    // Xre = C*Y're + dir*S*Y'im ; Xim = C*Y'im - dir*S*Y're
#pragma unroll
    for (int i = 0; i < 2; ++i)
#pragma unroll
      for (int j = 0; j < 2; ++j) {
        cre[i][j] = wmma_f32(aC[i],  bRe[j], cre[i][j]);
        cre[i][j] = wmma_f32(aSp[i], bIm[j], cre[i][j]);
        cim[i][j] = wmma_f32(aC[i],  bIm[j], cim[i][j]);
        cim[i][j] = wmma_f32(aSm[i], bRe[j], cim[i][j]);
      }
  }

  // ---------------- Epilogue: X_mat[k1][k2] -> x[k2 + 32*k1] ---------------
#pragma unroll
  for (int i = 0; i < 2; ++i)
#pragma unroll
    for (int j = 0; j < 2; ++j)
#pragma unroll
      for (int r = 0; r < 8; ++r) {
        const int m = 16 * i + r + 8 * h;   // k1
        const int n = 16 * j + l16;         // k2
        re[n + 32 * m] = scale * cre[i][j][r];
        im[n + 32 * m] = scale * cim[i][j][r];
      }
}

// Build the 32x32 DFT cos/sin tables in LDS (block-cooperative).
static __device__ __forceinline__ void init_tables(float* sC, float* sS,
                                                   int tid, int nthreads) {
  for (int e = tid; e < 1024; e += nthreads) {
    const int k = e >> 5, n = e & 31;
    const float ang = (2.0f * PI_F / 32.0f) * (float)((k * n) & 31);
    sC[e] = __cosf(ang);
    sS[e] = __sinf(ang);
  }
}

// ---------------------------------------------------------------------------
// K1: rows, forward.  out[n][r][:] = fft1024( H[n][r][:] * W[r][:] )
// Block = 128 threads = 4 waves, 4 rows per block.
// ---------------------------------------------------------------------------
__global__ __launch_bounds__(128) void k1_row_fwd(
    const float2* __restrict__ H, const float2* __restrict__ W,
    float2* __restrict__ out)
{
  __shared__ float sC[1024], sS[1024];
  __shared__ float ln[4][2][1024];

  const int tid = threadIdx.x;
  init_tables(sC, sS, tid, 128);
  __syncthreads();

  const int lane = tid & 31, w = tid >> 5;
  const int r = blockIdx.x * 4 + w;
  const size_t rowBase = ((size_t)blockIdx.y * 1024 + r) * 1024;
  float* re = ln[w][0];
  float* im = ln[w][1];

  for (int t = lane; t < 1024; t += 32) {
    const float2 hv = H[rowBase + t];
    const float2 wv = W[(size_t)r * 1024 + t];
    re[t] = hv.x * wv.x - hv.y * wv.y;     // G = H * W (complex)
    im[t] = hv.x * wv.y + hv.y * wv.x;
  }
  wave_fft1024(re, im, sC, sS, 1.0f, 1.0f, lane);
  for (int t = lane; t < 1024; t += 32)
    out[rowBase + t] = make_float2(re[t], im[t]);
}

// ---------------------------------------------------------------------------
// K2: columns, in-place.  S[:,v] = ifft1024( (fft1024(S[:,v]) - B[:,v])
//                                            * conj(T[:,v]) )
// Block = 128 threads = 4 waves; 4-column LDS tile (padded: no bank conflicts).
// ---------------------------------------------------------------------------
__global__ __launch_bounds__(128) void k2_col(
    float2* __restrict__ S, const float* __restrict__ Q,
    const float2* __restrict__ B, const float* __restrict__ z)
{
  __shared__ float sC[1024], sS[1024];
  __shared__ float colRe[4][1025], colIm[4][1025];

  const int tid = threadIdx.x;
  init_tables(sC, sS, tid, 128);

  const int c0 = blockIdx.x * 4;
  const size_t imgBase = (size_t)blockIdx.y * 1024 * 1024;
  const float zc = z[0] / C_LAMBDA_F;

  // Cooperative coalesced tile load (1024 rows x 4 cols).
  for (int e = tid; e < 4096; e += 128) {
    const int rr = e >> 2, cc = e & 3;
    const float2 v = S[imgBase + (size_t)rr * 1024 + c0 + cc];
    colRe[cc][rr] = v.x;
    colIm[cc][rr] = v.y;
  }
  __syncthreads();

  const int lane = tid & 31, w = tid >> 5;
  wave_fft1024(colRe[w], colIm[w], sC, sS, 1.0f, 1.0f, lane);  // forward
  __syncthreads();

  // Pointwise: (X - B) * conj(T), cooperative over the tile (coalesced in v).
  for (int e = tid; e < 4096; e += 128) {
    const int u = e >> 2, cc = e & 3;
    const int v = c0 + cc;
    const float ph = zc * Q[(size_t)u * 1024 + v];
    float ts, tc;
    __sincosf(ph, &ts, &tc);                 // T = tc + i*ts; conj: tc - i*ts
    const float2 b = B[(size_t)u * 1024 + v];
    const float xr = colRe[cc][u] - b.x;
    const float xi = colIm[cc][u] - b.y;
    colRe[cc][u] = xr * tc + xi * ts;
    colIm[cc][u] = xi * tc - xr * ts;
  }
  __syncthreads();

  wave_fft1024(colRe[w], colIm[w], sC, sS, -1.0f, 1.0f, lane); // inverse
  __syncthreads();

  // Cooperative coalesced tile store (in place).
  for (int e = tid; e < 4096; e += 128) {
    const int rr = e >> 2, cc = e & 3;
    S[imgBase + (size_t)rr * 1024 + c0 + cc] =
        make_float2(colRe[cc][rr], colIm[cc][rr]);
  }
}

// ---------------------------------------------------------------------------
// K3: rows, inverse, in-place, with total 2D-ifft scale 1/2^20.
// ---------------------------------------------------------------------------
__global__ __launch_bounds__(128) void k3_row_inv(float2* __restrict__ S)
{
  __shared__ float sC[1024], sS[1024];
  __shared__ float ln[4][2][1024];

  const int tid = threadIdx.x;
  init_tables(sC, sS, tid, 128);
  __syncthreads();

  const int lane = tid & 31, w = tid >> 5;
  const int r = blockIdx.x * 4 + w;
  const size_t rowBase = ((size_t)blockIdx.y * 1024 + r) * 1024;
  float* re = ln[w][0];
  float* im = ln[w][1];

  for (int t = lane; t < 1024; t += 32) {
    const float2 v = S[rowBase + t];
    re[t] = v.x;
    im[t] = v.y;
  }
  wave_fft1024(re, im, sC, sS, -1.0f, 1.0f / 1048576.0f, lane);
  for (int t = lane; t < 1024; t += 32)
    S[rowBase + t] = make_float2(re[t], im[t]);
}

// ---------------------------------------------------------------------------
extern "C" void kernel_launch(void* const* d_in, const int* in_sizes, int n_in,
                              void* d_out, int out_size, void* d_ws,
                              size_t ws_size, hipStream_t stream) {
  (void)in_sizes; (void)n_in; (void)out_size; (void)d_ws; (void)ws_size;
  const float2* H = (const float2*)d_in[0];  // [32,1024,1024,2] f32
  const float*  Q = (const float*) d_in[1];  // [1024,1024] f32
  const float*  z = (const float*) d_in[2];  // [1] f32
  const float2* W = (const float2*)d_in[3];  // [1024,1024,2] f32
  const float2* B = (const float2*)d_in[4];  // [1024,1024,2] f32
  float2* out = (float2*)d_out;              // used as the inter-pass scratch

  const dim3 blk(128);
  k1_row_fwd<<<dim3(256, 32), blk, 0, stream>>>(H, W, out);
  k2_col    <<<dim3(256, 32), blk, 0, stream>>>(out, Q, B, z);
  k3_row_inv<<<dim3(256, 32), blk, 0, stream>>>(out);
}